// EvolveGCN_87342454931925
// MI455X (gfx1250) — compile-verified
//
#include <hip/hip_runtime.h>
#include <cstdint>
#include <cstddef>

// Problem constants (from the reference)
#define TNUM   6
#define NNODES 50000
#define ENUM   800000
#define FDIM   128
#define LNUM   2
#define NEG_SLOPE 0.22916666666666666f  // (1/8 + 1/3)/2

typedef float v2f __attribute__((ext_vector_type(2)));
typedef float v8f __attribute__((ext_vector_type(8)));

#if defined(__has_builtin)
#if __has_builtin(__builtin_amdgcn_wmma_f32_16x16x4_f32)
#define USE_WMMA_F32 1
#endif
#endif

// ---------------------------------------------------------------------------
// One wave computes a 16x16 tile of C += A(128-deep K) * B using fp32 WMMA.
// A is accessed through ael(row,k), B through bel(k,col) so callers can fuse
// (gW+gU), (R.*Q), etc. for free.
//
// VGPR layouts per CDNA5 ISA 7.12.2:
//   A 16x4 f32 : lanes 0-15 -> M=lane, VGPR0=K0,VGPR1=K1 ; lanes 16-31 -> K2,K3
//   B  4x16 f32: mirrored (lane -> N, same K split)
//   C/D 16x16  : VGPR i -> M = row0 + i + 8*(lane>=16), N = col0 + (lane&15)
// ---------------------------------------------------------------------------
template <typename FA, typename FB>
__device__ inline void tile_gemm128(v8f& acc, FA ael, FB bel,
                                    int row0, int col0, int lane)
{
#ifdef USE_WMMA_F32
    const int h  = lane >> 4;
    const int ln = lane & 15;
    const int m  = row0 + ln;
    const int n  = col0 + ln;
    for (int k0 = 0; k0 < FDIM; k0 += 4) {
        const int kA = k0 + 2 * h;
        v2f a; a.x = ael(m, kA);  a.y = ael(m, kA + 1);
        v2f b; b.x = bel(kA, n);  b.y = bel(kA + 1, n);
        acc = __builtin_amdgcn_wmma_f32_16x16x4_f32(
            /*neg_a=*/false, a, /*neg_b=*/false, b,
            /*c_mod=*/(short)0, acc, /*reuse_a=*/false, /*reuse_b=*/false);
    }
#else
    // Scalar fallback producing identical logical placement in acc[].
    const int h  = lane >> 4;
    const int ln = lane & 15;
    const int n  = col0 + ln;
    for (int i = 0; i < 8; ++i) {
        const int r = row0 + i + 8 * h;
        float s = acc[i];
        for (int k = 0; k < FDIM; ++k) s += ael(r, k) * bel(k, n);
        acc[i] = s;
    }
#endif
}

__device__ inline float sigmoidf_(float v) { return 1.0f / (1.0f + expf(-v)); }

// ---------------------------------------------------------------------------
// GRU part 1: Z = sigmoid((gW0+gU0)@Q + b0), R = sigmoid((gW1+gU1)@Q + b1)
// 128 tiles total (2 gates x 64 tiles) -> 16 blocks x 8 waves.
// ---------------------------------------------------------------------------
__global__ __launch_bounds__(256) void gru_zr_kernel(
    const float* __restrict__ Q,
    const float* __restrict__ gW, const float* __restrict__ gU,
    const float* __restrict__ gb,
    float* __restrict__ Z, float* __restrict__ R)
{
    const int lane = threadIdx.x & 31;
    const int wave = threadIdx.x >> 5;
    const int id   = blockIdx.x * 8 + wave;   // 0..127
    const int gate = id >> 6;                 // 0 = z, 1 = r
    const int tile = id & 63;
    const int row0 = (tile >> 3) * 16;
    const int col0 = (tile & 7) * 16;

    const size_t FF = (size_t)FDIM * FDIM;
    const float* A0 = gW + (size_t)gate * FF;
    const float* A1 = gU + (size_t)gate * FF;
    const float* bias = gb + (size_t)gate * FF;

    const int h  = lane >> 4;
    const int n  = col0 + (lane & 15);

    v8f acc;
    for (int i = 0; i < 8; ++i)
        acc[i] = bias[(size_t)(row0 + i + 8 * h) * FDIM + n];

    tile_gemm128(acc,
        [&](int r, int k) { const size_t o = (size_t)r * FDIM + k;
                            return A0[o] + A1[o]; },
        [&](int k, int c) { return Q[(size_t)k * FDIM + c]; },
        row0, col0, lane);

    float* dst = (gate == 0) ? Z : R;
    for (int i = 0; i < 8; ++i)
        dst[(size_t)(row0 + i + 8 * h) * FDIM + n] = sigmoidf_(acc[i]);
}

// ---------------------------------------------------------------------------
// GRU part 2: Hh = tanh(gW2@Q + gU2@(R.*Q) + b2); Qnext = (1-Z).*Q + Z.*Hh
// 64 tiles -> 8 blocks x 8 waves.
// ---------------------------------------------------------------------------
__global__ __launch_bounds__(256) void gru_h_kernel(
    const float* __restrict__ Q,
    const float* __restrict__ gW2, const float* __restrict__ gU2,
    const float* __restrict__ gb2,
    const float* __restrict__ Z, const float* __restrict__ R,
    float* __restrict__ Qnext)
{
    const int lane = threadIdx.x & 31;
    const int wave = threadIdx.x >> 5;
    const int tile = blockIdx.x * 8 + wave;   // 0..63
    const int row0 = (tile >> 3) * 16;
    const int col0 = (tile & 7) * 16;

    const int h = lane >> 4;
    const int n = col0 + (lane & 15);

    v8f acc;
    for (int i = 0; i < 8; ++i)
        acc[i] = gb2[(size_t)(row0 + i + 8 * h) * FDIM + n];

    tile_gemm128(acc,
        [&](int r, int k) { return gW2[(size_t)r * FDIM + k]; },
        [&](int k, int c) { return Q[(size_t)k * FDIM + c]; },
        row0, col0, lane);

    tile_gemm128(acc,
        [&](int r, int k) { return gU2[(size_t)r * FDIM + k]; },
        [&](int k, int c) { const size_t o = (size_t)k * FDIM + c;
                            return R[o] * Q[o]; },
        row0, col0, lane);

    for (int i = 0; i < 8; ++i) {
        const size_t o = (size_t)(row0 + i + 8 * h) * FDIM + n;
        const float hv = tanhf(acc[i]);
        const float q  = Q[o];
        const float z  = Z[o];
        Qnext[o] = (1.0f - z) * q + z * hv;
    }
}

// ---------------------------------------------------------------------------
// H = X(t) @ Q :  (N x 128) @ (128 x 128).  One 16x16 tile per wave.
// grid = N/16 blocks, 8 waves per block cover the 8 column tiles.
// ---------------------------------------------------------------------------
__global__ __launch_bounds__(256) void hgemm_kernel(
    const float* __restrict__ X, const float* __restrict__ Q,
    float* __restrict__ H)
{
    const int lane = threadIdx.x & 31;
    const int wave = threadIdx.x >> 5;
    const int row0 = blockIdx.x * 16;
    const int col0 = wave * 16;

    const int h = lane >> 4;
    const int n = col0 + (lane & 15);

    v8f acc;
    for (int i = 0; i < 8; ++i) acc[i] = 0.0f;

    tile_gemm128(acc,
        [&](int r, int k) { return X[(size_t)r * FDIM + k]; },
        [&](int k, int c) { return Q[(size_t)k * FDIM + c]; },
        row0, col0, lane);

    for (int i = 0; i < 8; ++i)
        H[(size_t)(row0 + i + 8 * h) * FDIM + n] = acc[i];
}

// ---------------------------------------------------------------------------
// agg[dst[e], :] += w[e] * H[col[e], :]
// One wave per edge; float4 per lane covers all 128 features.
// Edge metadata loads are forced wave-uniform (readfirstlane) so the backend
// can emit scalar s_load_b64/b32 instead of 32 redundant vector loads.
// The next edge's H row is prefetched (global_prefetch_b8) to hide gather
// latency behind the current edge's atomics.
// ---------------------------------------------------------------------------
__global__ __launch_bounds__(256) void scatter_kernel(
    const float* __restrict__ H,
    const long long* __restrict__ dstIdx,
    const long long* __restrict__ colIdx,
    const float* __restrict__ ew,
    float* __restrict__ out)
{
    const int lane = threadIdx.x & 31;
    const int gw   = (int)((blockIdx.x * blockDim.x + threadIdx.x) >> 5);
    const int nw   = (int)((gridDim.x * blockDim.x) >> 5);

    int e = __builtin_amdgcn_readfirstlane(gw);        // wave-uniform SGPR
    const int stride = __builtin_amdgcn_readfirstlane(nw);

    while (e < ENUM) {
        const int   s = (int)colIdx[e];
        const int   d = (int)dstIdx[e];
        const float w = ew[e];

        const int en = e + stride;                     // uniform branch
        if (en < ENUM) {
            const int sn = (int)colIdx[en];
            __builtin_prefetch(H + (size_t)sn * FDIM + lane * 4, 0, 3);
        }

        const float4 hv = *(const float4*)(H + (size_t)s * FDIM + lane * 4);
        float* o = out + (size_t)d * FDIM + lane * 4;
        atomicAdd(o + 0, w * hv.x);
        atomicAdd(o + 1, w * hv.y);
        atomicAdd(o + 2, w * hv.z);
        atomicAdd(o + 3, w * hv.w);

        e = en;
    }
}

__global__ void zero4_kernel(float4* __restrict__ p, int n4)
{
    const int i = blockIdx.x * blockDim.x + threadIdx.x;
    if (i < n4) p[i] = make_float4(0.f, 0.f, 0.f, 0.f);
}

__global__ void rrelu4_kernel(float4* __restrict__ p, int n4)
{
    const int i = blockIdx.x * blockDim.x + threadIdx.x;
    if (i < n4) {
        float4 v = p[i];
        v.x = (v.x >= 0.f) ? v.x : NEG_SLOPE * v.x;
        v.y = (v.y >= 0.f) ? v.y : NEG_SLOPE * v.y;
        v.z = (v.z >= 0.f) ? v.z : NEG_SLOPE * v.z;
        v.w = (v.w >= 0.f) ? v.w : NEG_SLOPE * v.w;
        p[i] = v;
    }
}

// ---------------------------------------------------------------------------
extern "C" void kernel_launch(void* const* d_in, const int* in_sizes, int n_in,
                              void* d_out, int out_size, void* d_ws, size_t ws_size,
                              hipStream_t stream)
{
    (void)in_sizes; (void)n_in; (void)out_size; (void)ws_size;

    const float*     x   = (const float*)d_in[0];      // (T,N,F)
    const long long* ei  = (const long long*)d_in[1];  // (T,2,E) int64
    const float*     ew  = (const float*)d_in[2];      // (T,E)
    const float*     gWa = (const float*)d_in[3];      // (L,3,F,F)
    const float*     gUa = (const float*)d_in[4];      // (L,3,F,F)
    const float*     gba = (const float*)d_in[5];      // (L,3,F,F)
    const float*     W0  = (const float*)d_in[6];      // (L,F,F)
    float*           out = (float*)d_out;              // (T,N,F)

    const size_t FF  = (size_t)FDIM * FDIM;
    const size_t NF  = (size_t)NNODES * FDIM;

    char*  ws   = (char*)d_ws;
    float* Hbuf = (float*)ws;  ws += NF * sizeof(float);           // 25.6 MB
    float* Qa   = (float*)ws;  ws += FF * sizeof(float);
    float* Qb   = (float*)ws;  ws += FF * sizeof(float);
    float* Zb   = (float*)ws;  ws += FF * sizeof(float);
    float* Rb   = (float*)ws;  ws += FF * sizeof(float);

    const int n4       = (int)(NF / 4);                 // 1,600,000
    const int ewBlocks = (n4 + 255) / 256;              // 6250

    for (int l = 0; l < LNUM; ++l) {
        const float* xin = (l == 0) ? x : out;          // layer input (T,N,F)
        const float* lgW = gWa + (size_t)l * 3 * FF;
        const float* lgU = gUa + (size_t)l * 3 * FF;
        const float* lgb = gba + (size_t)l * 3 * FF;

        // Q0 = W0[l]
        hipMemcpyAsync(Qa, W0 + (size_t)l * FF, FF * sizeof(float),
                       hipMemcpyDeviceToDevice, stream);
        float* qcur = Qa;
        float* qnxt = Qb;

        for (int t = 0; t < TNUM; ++t) {
            // Q = GRU(Q)
            gru_zr_kernel<<<16, 256, 0, stream>>>(qcur, lgW, lgU, lgb, Zb, Rb);
            gru_h_kernel<<<8, 256, 0, stream>>>(qcur, lgW + 2 * FF, lgU + 2 * FF,
                                                lgb + 2 * FF, Zb, Rb, qnxt);

            // H = x_t @ Q   (reads layer input slice BEFORE we overwrite it)
            hgemm_kernel<<<NNODES / 16, 256, 0, stream>>>(
                xin + (size_t)t * NF, qnxt, Hbuf);

            // out[t] = rrelu(segment_sum(w .* H[col], dst))
            float* ot = out + (size_t)t * NF;
            zero4_kernel<<<ewBlocks, 256, 0, stream>>>((float4*)ot, n4);
            const long long* dstIdx = ei + (size_t)t * 2 * ENUM;        // row 0
            const long long* colIdx = dstIdx + ENUM;                    // row 1
            scatter_kernel<<<2048, 256, 0, stream>>>(
                Hbuf, dstIdx, colIdx, ew + (size_t)t * ENUM, ot);
            rrelu4_kernel<<<ewBlocks, 256, 0, stream>>>((float4*)ot, n4);

            float* tmp = qcur; qcur = qnxt; qnxt = tmp;  // ping-pong Q
        }
    }
}